// Coocurrence_28226525069525
// MI455X (gfx1250) — compile-verified
//
#include <hip/hip_runtime.h>

// Co-occurrence scatter-add:
//   out[l, r] += 1.0f for each of N_PAIRS (l, r) pairs, on top of input weight.
// Memory/atomic-bound workload. Optimal MI455X path:
//   pass 1: b128 copy weight -> out (~512 MB @ 23.3 TB/s ~ 22 us)
//   pass 2: streamed NT b128 index loads + non-returning GLOBAL_ATOMIC_ADD_F32
//           into a mostly-L2-resident (192 MB) 256 MB target.
// No dense-matmul structure exists here, so WMMA does not apply.

#define NV 8192  // N_VOCAB

typedef int   int4v   __attribute__((ext_vector_type(4)));
typedef float float4v __attribute__((ext_vector_type(4)));

// Pass 2 (defined first so the disasm snippet shows the atomic lowering):
// 4 pairs per thread. Index arrays are streamed exactly once -> non-temporal
// b128 loads. Atomic return value discarded -> non-returning
// global_atomic_add_f32 (STOREcnt), executed at the L2 atomic units.
__global__ __launch_bounds__(256) void cooc_scatter(const int* __restrict__ left,
                                                    const int* __restrict__ right,
                                                    float* __restrict__ out,
                                                    int n) {
  const float one = 1.0f;
  int base = (blockIdx.x * 256 + threadIdx.x) * 4;
  if (base + 4 <= n) {
    int4v l = __builtin_nontemporal_load((const int4v*)(left + base));
    int4v r = __builtin_nontemporal_load((const int4v*)(right + base));
    // row*8192 + col fits in 32 bits (max 8191*8192+8191 = 0x3FFFFFF).
    unsafeAtomicAdd(&out[((unsigned)l.x << 13) + (unsigned)r.x], one);
    unsafeAtomicAdd(&out[((unsigned)l.y << 13) + (unsigned)r.y], one);
    unsafeAtomicAdd(&out[((unsigned)l.z << 13) + (unsigned)r.z], one);
    unsafeAtomicAdd(&out[((unsigned)l.w << 13) + (unsigned)r.w], one);
  } else if (base < n) {
    for (int i = base; i < n; ++i) {
      unsafeAtomicAdd(&out[((unsigned)left[i] << 13) + (unsigned)right[i]], one);
    }
  }
}

// Pass 1: materialize weight into d_out (d_out is poisoned by the harness).
// float4 copy -> global_load_b128 / global_store_b128, regular temporal so the
// destination lines are warm in L2 for the atomic phase.
__global__ __launch_bounds__(256) void cooc_init(const float4v* __restrict__ w,
                                                 float4v* __restrict__ out,
                                                 int n4) {
  int i = blockIdx.x * 256 + threadIdx.x;
  if (i < n4) {
    out[i] = w[i];
  }
}

extern "C" void kernel_launch(void* const* d_in, const int* in_sizes, int n_in,
                              void* d_out, int out_size, void* d_ws, size_t ws_size,
                              hipStream_t stream) {
  const int*   left   = (const int*)d_in[0];
  const int*   right  = (const int*)d_in[1];
  const float* weight = (const float*)d_in[2];
  float*       out    = (float*)d_out;

  const int n_pairs = in_sizes[0];

  // Pass 1: copy weight -> out as float4 (out_size = 8192*8192, divisible by 4).
  const int n4 = out_size >> 2;
  const int init_blocks = (n4 + 255) / 256;
  cooc_init<<<init_blocks, 256, 0, stream>>>(
      (const float4v*)weight, (float4v*)out, n4);

  // Pass 2: scatter-add, 4 pairs per thread (stream-ordered after pass 1).
  const int threads = (n_pairs + 3) / 4;
  const int scatter_blocks = (threads + 255) / 256;
  cooc_scatter<<<scatter_blocks, 256, 0, stream>>>(left, right, out, n_pairs);
}